// FECONV_14121852470122
// MI455X (gfx1250) — compile-verified
//
#include <hip/hip_runtime.h>

typedef float v2f __attribute__((ext_vector_type(2)));
typedef float v8f __attribute__((ext_vector_type(8)));

#define WAVES_PER_BLOCK 8
#define NTYPES 64

// ---------------- helper kernels ----------------

__global__ void fe_zero_f32(float* __restrict__ p, int n) {
  int i = blockIdx.x * 256 + threadIdx.x;
  if (i < n) p[i] = 0.0f;
}

__global__ void fe_init(int* __restrict__ counts, int* __restrict__ perm, int maxPadded) {
  int i = blockIdx.x * 256 + threadIdx.x;
  if (i < maxPadded) perm[i] = -1;
  if (i < NTYPES) counts[i] = 0;
}

__global__ void fe_hist(const int* __restrict__ types, int* __restrict__ counts, int E) {
  int e = blockIdx.x * 256 + threadIdx.x;
  if (e < E) atomicAdd(&counts[types[e]], 1);
}

__global__ void fe_scan(const int* __restrict__ counts, int* __restrict__ cursor) {
  if (threadIdx.x == 0 && blockIdx.x == 0) {
    int run = 0;
    for (int t = 0; t < NTYPES; ++t) {
      cursor[t] = run;
      run += (counts[t] + 15) & ~15;   // pad each bin to a multiple of 16
    }
  }
}

__global__ void fe_scatter(const int* __restrict__ types, int* __restrict__ cursor,
                           int* __restrict__ perm, int E) {
  int e = blockIdx.x * 256 + threadIdx.x;
  if (e < E) {
    int pos = atomicAdd(&cursor[types[e]], 1);
    perm[pos] = e;
  }
}

// Pre-pack WMMA B operands (K^T) into fragment-ready, zero-padded layout:
//   Bpack[((t*12 + 2*s + tile)*32 + lane)*2 + {0,1}]
// B (4x16 f32) layout: lane 0-15 -> N=lane, {K=4s,4s+1}; lane 16-31 -> N=lane-16, {K=4s+2,4s+3}
// tile selects output DOF columns n = tile*16 + (lane&15); n >= 24 is zero padding.
__global__ void fe_packB(const float* __restrict__ filters, float* __restrict__ Bpack, int nTypes) {
  int idx = blockIdx.x * 256 + threadIdx.x;     // (t*12 + frag)*32 + lane
  int total = nTypes * 12 * 32;
  if (idx >= total) return;
  int lane = idx & 31;
  int frag = (idx >> 5) % 12;
  int t    = idx / (12 * 32);
  int s    = frag >> 1;
  int tile = frag & 1;
  int half = lane >> 4;
  int l15  = lane & 15;
  int n    = tile * 16 + l15;                   // output DOF column
  int j    = 4 * s + 2 * half;                  // K index (input DOF)
  float x = 0.0f, y = 0.0f;
  if (n < 24) {
    const float* row = filters + (size_t)t * 576 + n * 24 + j;  // K^T[j][n] = K[n][j]
    x = row[0]; y = row[1];
  }
  float* dst = Bpack + (size_t)idx * 2;
  dst[0] = x; dst[1] = y;
}

// ---------------- main compute: one wave == 16 same-type elements ----------------
//
// F[16,24] = Ue[16,24] * K^T  via V_WMMA_F32_16X16X4_F32:
//   A (16x4 f32): lane 0-15 -> M=lane, {K=4s,4s+1}; lane 16-31 -> M=lane-16, {K=4s+2,4s+3}
//   D (16x16 f32): VGPR r -> M = r + 8*(lane>=16), N = lane&15

__global__ __launch_bounds__(256) void fe_apply(
    const float* __restrict__ U, const float* __restrict__ Bpack,
    const int* __restrict__ types, const int* __restrict__ nodIdx,
    const int* __restrict__ perm, float* __restrict__ KU, int maxPadded)
{
  __shared__ float sUe[WAVES_PER_BLOCK][16][25];  // stride 25 to dodge bank conflicts
  __shared__ int   sNod[WAVES_PER_BLOCK][16][8];  // cached connectivity
  __shared__ int   sEid[WAVES_PER_BLOCK][16];

  const int lane  = threadIdx.x & 31;
  const int wv    = threadIdx.x >> 5;
  const int group = blockIdx.x * WAVES_PER_BLOCK + wv;
  const int base  = group * 16;

  // fetch the 16 element ids for this wave's group
  if (lane < 16) {
    int eid = -1;
    int p = base + lane;
    if (p < maxPadded) eid = perm[p];
    sEid[wv][lane] = eid;
  }
  __syncthreads();

  // stage Ue[16][24] + connectivity into LDS: 128 (elem,node) slots, 4 per lane
  #pragma unroll
  for (int it = 0; it < 4; ++it) {
    int slot = it * 32 + lane;
    int m  = slot >> 3;
    int nd = slot & 7;
    int e  = sEid[wv][m];
    float x = 0.0f, y = 0.0f, z = 0.0f;
    int node = 0;
    if (e >= 0) {
      node = nodIdx[(size_t)e * 8 + nd];
      const float* u = U + (size_t)node * 3;
      x = u[0]; y = u[1]; z = u[2];
    }
    sNod[wv][m][nd] = node;
    float* dst = &sUe[wv][m][nd * 3];
    dst[0] = x; dst[1] = y; dst[2] = z;
  }
  __syncthreads();

  const int e0 = sEid[wv][0];             // wave-uniform; padding is at bin tails
  if (e0 < 0) return;                     // fully padded group (no barriers below)

  const int t    = types[e0];
  const int half = lane >> 4;             // 0 or 1
  const int l15  = lane & 15;

  // fragment-ready B operands for this type: v2f per lane per fragment
  const v2f* bfrag = (const v2f*)Bpack + ((size_t)t * 12) * 32 + lane;

  v8f c0 = {0,0,0,0,0,0,0,0};
  v8f c1 = {0,0,0,0,0,0,0,0};

  #pragma unroll
  for (int s = 0; s < 6; ++s) {
    const int j = 4 * s + 2 * half;
    v2f a;  a.x = sUe[wv][l15][j];  a.y = sUe[wv][l15][j + 1];
    v2f b0 = bfrag[(2 * s + 0) * 32];
    v2f b1 = bfrag[(2 * s + 1) * 32];
    c0 = __builtin_amdgcn_wmma_f32_16x16x4_f32(false, a, false, b0, (short)0, c0, false, false);
    c1 = __builtin_amdgcn_wmma_f32_16x16x4_f32(false, a, false, b1, (short)0, c1, false, false);
  }

  // scatter-add: lane owns DOF columns i0 (tile0) and i1 (tile1), VGPR r owns element m
  const int i0   = l15;
  const int i1   = 16 + l15;
  const int i0d3 = i0 / 3, i0m3 = i0 % 3;
  const int i1d3 = i1 / 3, i1m3 = i1 % 3;
  #pragma unroll
  for (int r = 0; r < 8; ++r) {
    int m = r + 8 * half;
    int e = sEid[wv][m];
    if (e >= 0) {
      int nA = sNod[wv][m][i0d3];
      atomicAdd(&KU[(size_t)nA * 3 + i0m3], c0[r]);
      if (i1 < 24) {
        int nB = sNod[wv][m][i1d3];
        atomicAdd(&KU[(size_t)nB * 3 + i1m3], c1[r]);
      }
    }
  }
}

// ---------------- launcher ----------------

extern "C" void kernel_launch(void* const* d_in, const int* in_sizes, int n_in,
                              void* d_out, int out_size, void* d_ws, size_t ws_size,
                              hipStream_t stream) {
  const float* U       = (const float*)d_in[0];
  const float* filters = (const float*)d_in[1];
  const int*   types   = (const int*)d_in[2];
  const int*   nodIdx  = (const int*)d_in[3];
  float*       KU      = (float*)d_out;

  const int E         = in_sizes[2];              // number of elements
  const int T         = in_sizes[1] / 576;        // number of stiffness types
  const int maxPadded = E + NTYPES * 16;          // worst-case padded perm length

  int*   counts = (int*)d_ws;                     // [64]
  int*   cursor = counts + NTYPES;                // [64]
  float* Bpack  = (float*)(counts + 2 * NTYPES);  // [T*12*32*2] fragment-packed K^T
  int*   perm   = (int*)(Bpack + (size_t)T * 768);// [maxPadded]

  fe_zero_f32<<<(out_size + 255) / 256, 256, 0, stream>>>(KU, out_size);
  fe_init   <<<(maxPadded + 255) / 256, 256, 0, stream>>>(counts, perm, maxPadded);
  fe_hist   <<<(E + 255) / 256, 256, 0, stream>>>(types, counts, E);
  fe_scan   <<<1, 64, 0, stream>>>(counts, cursor);
  fe_scatter<<<(E + 255) / 256, 256, 0, stream>>>(types, cursor, perm, E);
  fe_packB  <<<(T * 12 * 32 + 255) / 256, 256, 0, stream>>>(filters, Bpack, T);

  const int groups = (maxPadded + 15) / 16;
  const int blocks = (groups + WAVES_PER_BLOCK - 1) / WAVES_PER_BLOCK;
  fe_apply  <<<blocks, 256, 0, stream>>>(U, Bpack, types, nodIdx, perm, KU, maxPadded);
}